// KernelFilterClass_46823733461000
// MI455X (gfx1250) — compile-verified
//
#include <hip/hip_runtime.h>

#define AS1 __attribute__((address_space(1)))
#define AS3 __attribute__((address_space(3)))
typedef int v4i __attribute__((__vector_size__(16)));

// Problem constants (from reference): B=8, C=32, H=W=512, K=7, dilation=2
constexpr int Bn = 8, Cn = 32, Hn = 512, Wn = 512;
constexpr int KT = 7, K2 = 49, PADR = 6;   // pad = (K/2)*dilation = 6
constexpr int WT   = 64;        // output pixels per block tile (one h row)
constexpr int GROW = 80;        // padded LDS grid row: covers w0-8 .. w0+71
constexpr int NCG  = 4;         // channels in flight per LDS stage
constexpr int NCO  = Cn / NCG;  // 8 channel-group iterations
constexpr int GCHUNKS = NCG * KT * (GROW / 4);   // 560 16-B chunks per stage
constexpr int KCHUNKS = (K2 * WT) / 4;           // 784 16-B chunks of taps

__device__ __forceinline__ void async_b128(const float* g, float* l) {
#if __has_builtin(__builtin_amdgcn_global_load_async_to_lds_b128)
    __builtin_amdgcn_global_load_async_to_lds_b128(
        (AS1 v4i*)g, (AS3 v4i*)l, /*offset=*/0, /*cpol=*/0);
#else
    *(float4*)l = *(const float4*)g;
#endif
}

__device__ __forceinline__ void wait_async0() {
#if __has_builtin(__builtin_amdgcn_s_wait_asynccnt)
    __builtin_amdgcn_s_wait_asynccnt(0);
#else
    asm volatile("s_wait_asynccnt 0" ::: "memory");
#endif
}

__global__ __launch_bounds__(256)
void KernelFilter_cdna5(const float* __restrict__ grid,
                        const float* __restrict__ kern,
                        float* __restrict__ out)
{
    __shared__ __align__(16) float lds_k[K2 * WT];            // 12544 B taps
    __shared__ __align__(16) float lds_g[2][NCG * KT * GROW]; // 2 x 8960 B grid

    const int tid = threadIdx.x;
    const int px  = tid & (WT - 1);   // pixel within tile
    const int cg  = tid >> 6;         // 0..3 channel sub-group
    const int w0  = blockIdx.x * WT;
    const int h   = blockIdx.y;
    const int b   = blockIdx.z;

    const size_t plane = (size_t)Hn * Wn;

    // Async-stage 4 channels x 7 haloed rows x 80 floats into lds_g[buf].
    // Every 16-B chunk is fully in- or fully out-of-image (W,w0,cols all 4-aligned).
    auto stage_grid = [&](int co, int buf) {
        for (int q = tid; q < GCHUNKS; q += 256) {
            const int cgi = q / (KT * 20);           // channel sub-group
            const int rem = q - cgi * (KT * 20);
            const int i   = rem / 20;                // vertical tap row
            const int k4  = rem - i * 20;            // chunk within row
            const int c   = co * NCG + cgi;
            const int hs  = h + 2 * i - PADR;
            const int ws  = w0 - 8 + 4 * k4;         // always 4-aligned
            float* l = &lds_g[buf][(cgi * KT + i) * GROW + 4 * k4];
            if ((unsigned)hs < (unsigned)Hn && (unsigned)ws < (unsigned)Wn) {
                const float* g =
                    grid + (size_t)(b * Cn + c) * plane + (size_t)hs * Wn + ws;
                async_b128(g, l);
            } else {
                *(float4*)l = make_float4(0.f, 0.f, 0.f, 0.f);  // zero halo
            }
        }
    };

    // ---- Prologue: DMA the 49x64 per-pixel tap tile + first grid stage.
    const float* kbase = kern + ((size_t)b * K2 * Hn + h) * Wn;
    for (int q = tid; q < KCHUNKS; q += 256) {
        const int t  = q >> 4;            // tap 0..48
        const int c4 = (q & 15) << 2;     // float offset within 64-px row
        async_b128(kbase + (size_t)t * plane + w0 + c4, &lds_k[t * WT + c4]);
    }
    stage_grid(0, 0);
    wait_async0();
    __syncthreads();

    // Each thread keeps its pixel's 49 taps in VGPRs: 8x reuse across channels.
    float kreg[K2];
#pragma unroll
    for (int t = 0; t < K2; ++t) kreg[t] = lds_k[t * WT + px];

    for (int co = 0; co < NCO; ++co) {
        const int buf = co & 1;
        // Overlap next stage's DMA with this stage's compute.
        if (co + 1 < NCO) stage_grid(co + 1, buf ^ 1);

        const float* gr = &lds_g[buf][cg * (KT * GROW)];
        float a0 = 0.f, a1 = 0.f, a2 = 0.f, a3 = 0.f;
#pragma unroll
        for (int i = 0; i < KT; ++i) {
#pragma unroll
            for (int j = 0; j < KT; ++j) {
                // tap (i,j) needs grid col (w0 + px + 2j - 6) -> LDS col px+2j+2
                const float t = kreg[i * 7 + j] * gr[i * GROW + px + 2 * j + 2];
                switch ((i * 7 + j) & 3) {   // 4 accumulators for FMA ILP
                    case 0: a0 += t; break;
                    case 1: a1 += t; break;
                    case 2: a2 += t; break;
                    default: a3 += t; break;
                }
            }
        }
        const int c = co * NCG + cg;
        out[(size_t)(b * Cn + c) * plane + (size_t)h * Wn + w0 + px] =
            (a0 + a1) + (a2 + a3);

        wait_async0();      // next buffer's DMA has overlapped the compute
        __syncthreads();
    }
}

extern "C" void kernel_launch(void* const* d_in, const int* in_sizes, int n_in,
                              void* d_out, int out_size, void* d_ws, size_t ws_size,
                              hipStream_t stream) {
    (void)in_sizes; (void)n_in; (void)d_ws; (void)ws_size; (void)out_size;
    const float* grid = (const float*)d_in[0];
    const float* kern = (const float*)d_in[1];
    // d_in[2] is dilation (== 2, baked into PADR and the 2*i/2*j strides)
    float* out = (float*)d_out;

    dim3 g(Wn / WT, Hn, Bn);   // (8, 512, 8) blocks
    dim3 blk(256);             // 8 wave32s
    KernelFilter_cdna5<<<g, blk, 0, stream>>>(grid, kern, out);
}